// MultiHeadAttentionLayer_13426067768106
// MI455X (gfx1250) — compile-verified
//
#include <hip/hip_runtime.h>
#include <math.h>

#define EMBED  128
#define NHEADS 4
#define HDIM   32
#define LSTRIDE 132              // padded LDS row stride (floats): bank-conflict-free, 16B aligned

typedef __attribute__((ext_vector_type(2))) float v2f;
typedef __attribute__((ext_vector_type(8))) float v8f;

// ---- monotone float <-> orderable uint (for atomic max over signed floats) ----
__device__ __forceinline__ unsigned f2ord(float f) {
    unsigned u = __float_as_uint(f);
    return (u & 0x80000000u) ? ~u : (u | 0x80000000u);
}
__device__ __forceinline__ float ord2f(unsigned u) {
    return (u & 0x80000000u) ? __uint_as_float(u & 0x7FFFFFFFu) : __uint_as_float(~u);
}

// ---------------------------------------------------------------------------
// 0) init: zero the scatter accumulator, init per-head max/sum
// ---------------------------------------------------------------------------
__global__ void init_kernel(float* __restrict__ accum, unsigned* __restrict__ hmaxo,
                            float* __restrict__ hsum, long long n) {
    long long i = (long long)blockIdx.x * blockDim.x + threadIdx.x;
    if (i < n) accum[i] = 0.0f;
    if (i < NHEADS) { hmaxo[i] = f2ord(-__builtin_inff()); hsum[i] = 0.0f; }
}

// ---------------------------------------------------------------------------
// GEMM core: one block = one 16-row slab of X @ W.  W (64KB) + A tile (8KB)
// staged in LDS (padded stride), 8 waves each own one 16-col tile.
// Inner loop: ds_load_b64 (A) + 2x ds_load_b32 (B) + v_wmma_f32_16x16x4_f32.
// ---------------------------------------------------------------------------
__device__ __forceinline__ void gemm_slab(const float* __restrict__ Xsrc,  // [N,128]
                                          const float* __restrict__ W,    // [128,128]
                                          float* __restrict__ Out,        // [N,128]
                                          const float* __restrict__ bias, // [128] or nullptr
                                          int rb, int N) {
    extern __shared__ float smem[];
    float* sW = smem;                      // 128 rows x LSTRIDE
    float* sA = smem + EMBED * LSTRIDE;    // 16 rows x LSTRIDE

    // cooperative fill: W (4096 float4 tiles), A (512 float4 tiles)
    for (int i = threadIdx.x; i < EMBED * (EMBED / 4); i += blockDim.x) {
        const int row = i >> 5, c4 = i & 31;
        *(float4*)(sW + row * LSTRIDE + c4 * 4) = *(const float4*)(W + row * EMBED + c4 * 4);
    }
    for (int i = threadIdx.x; i < 16 * (EMBED / 4); i += blockDim.x) {
        const int row = i >> 5, c4 = i & 31;
        const int grow = min(rb * 16 + row, N - 1);          // branch-free clamp
        *(float4*)(sA + row * LSTRIDE + c4 * 4) =
            *(const float4*)(Xsrc + (long long)grow * EMBED + c4 * 4);
    }
    __syncthreads();

    const int lane = threadIdx.x & 31;
    const int ct   = threadIdx.x >> 5;     // wave id = column tile 0..7
    const int half = lane >> 4;            // selects K pair {0,1} vs {2,3}
    const int l15  = lane & 15;
    const int bcol = ct * 16 + l15;

    v8f c = {0.f, 0.f, 0.f, 0.f, 0.f, 0.f, 0.f, 0.f};
    #pragma unroll 8
    for (int kk = 0; kk < 32; ++kk) {
        const int k0 = kk * 4 + half * 2;
        v2f a = *(const v2f*)(sA + l15 * LSTRIDE + k0);      // ds_load_b64
        v2f b;
        b.x = sW[(k0 + 0) * LSTRIDE + bcol];                 // ds_load_b32
        b.y = sW[(k0 + 1) * LSTRIDE + bcol];                 // ds_load_b32
        c = __builtin_amdgcn_wmma_f32_16x16x4_f32(false, a, false, b, (short)0, c, false, false);
    }

    const float bv = bias ? bias[bcol] : 0.0f;
    if (rb * 16 + 15 < N) {                                  // uniform (SGPR) branch
        #pragma unroll
        for (int j = 0; j < 8; ++j) {
            const int orow = rb * 16 + half * 8 + j;
            Out[(long long)orow * EMBED + bcol] = c[j] + bv;
        }
    } else {
        #pragma unroll
        for (int j = 0; j < 8; ++j) {
            const int orow = rb * 16 + half * 8 + j;
            if (orow < N) Out[(long long)orow * EMBED + bcol] = c[j] + bv;
        }
    }
}

// 1) QKV projections: gridDim.x = 3 * rowBlocks
__global__ void qkv_kernel(const float* __restrict__ X,
                           const float* __restrict__ Wq, const float* __restrict__ Wk,
                           const float* __restrict__ Wv,
                           float* __restrict__ q, float* __restrict__ k, float* __restrict__ v,
                           int N, int rowBlocks) {
    const int mat = blockIdx.x / rowBlocks;
    const int rb  = blockIdx.x - mat * rowBlocks;
    const float* W   = (mat == 0) ? Wq : (mat == 1) ? Wk : Wv;
    float*       Out = (mat == 0) ? q  : (mat == 1) ? k  : v;
    gemm_slab(X, W, Out, nullptr, rb, N);
}

// 5) output projection + bias: gridDim.x = rowBlocks
__global__ void outproj_kernel(const float* __restrict__ A, const float* __restrict__ Wo,
                               const float* __restrict__ bo, float* __restrict__ Out, int N) {
    gemm_slab(A, Wo, Out, bo, (int)blockIdx.x, N);
}

// ---------------------------------------------------------------------------
// 2) per-edge scores + global per-head max (q/k are L2-resident: 77MB < 192MB)
// ---------------------------------------------------------------------------
__global__ void score_kernel(const float* __restrict__ q, const float* __restrict__ k,
                             const long long* __restrict__ ei, float* __restrict__ scores,
                             unsigned* __restrict__ hmaxo, int E) {
    __shared__ unsigned smax[NHEADS];
    if (threadIdx.x < NHEADS) smax[threadIdx.x] = 0u;        // below f2ord(-inf)
    __syncthreads();
    const long long t = (long long)blockIdx.x * blockDim.x + threadIdx.x;
    if (t < (long long)E * NHEADS) {
        const int e = (int)(t >> 2);
        const int h = (int)(t & 3);
        const long long src = ei[e];
        const long long dst = ei[(long long)E + e];
        const float4* __restrict__ qp = (const float4*)(q + src * EMBED + h * HDIM);
        const float4* __restrict__ kp = (const float4*)(k + dst * EMBED + h * HDIM);
        float acc = 0.0f;
        #pragma unroll
        for (int i = 0; i < HDIM / 4; ++i) {
            float4 a = qp[i], b = kp[i];
            acc = fmaf(a.x, b.x, fmaf(a.y, b.y, fmaf(a.z, b.z, fmaf(a.w, b.w, acc))));
        }
        const float s = acc * 0.17677669529663687f;          // 1/sqrt(32)
        scores[t] = s;
        atomicMax(&smax[h], f2ord(s));
    }
    __syncthreads();
    if (threadIdx.x < NHEADS) atomicMax(&hmaxo[threadIdx.x], smax[threadIdx.x]);
}

// ---------------------------------------------------------------------------
// 3) exp(s - max) in place + global per-head sum
// ---------------------------------------------------------------------------
__global__ void expsum_kernel(float* __restrict__ scores, const unsigned* __restrict__ hmaxo,
                              float* __restrict__ hsum, int E) {
    __shared__ float ssum[NHEADS];
    if (threadIdx.x < NHEADS) ssum[threadIdx.x] = 0.0f;
    __syncthreads();
    const long long t = (long long)blockIdx.x * blockDim.x + threadIdx.x;
    if (t < (long long)E * NHEADS) {
        const int h = (int)(t & 3);
        const float ex = __expf(scores[t] - ord2f(hmaxo[h]));
        scores[t] = ex;
        atomicAdd(&ssum[h], ex);
    }
    __syncthreads();
    if (threadIdx.x < NHEADS) atomicAdd(&hsum[threadIdx.x], ssum[threadIdx.x]);
}

// ---------------------------------------------------------------------------
// 4) weighted scatter-add: accum[src] += v[dst] * attn  (L2-resident atomics)
//    one thread per (edge, 4-channel group): b128 load + 4x atomic_add_f32
// ---------------------------------------------------------------------------
__global__ void scatter_kernel(const float* __restrict__ scores, const float* __restrict__ hsum,
                               const float* __restrict__ v, const long long* __restrict__ ei,
                               float* __restrict__ accum, int E) {
    const long long t = (long long)blockIdx.x * blockDim.x + threadIdx.x;
    if (t >= (long long)E * (EMBED / 4)) return;
    const int e  = (int)(t >> 5);
    const int r4 = (int)(t & 31) * 4;                        // channel group 0..124
    const int h  = r4 >> 5;
    const float attn = scores[(long long)e * NHEADS + h] / hsum[h];
    const long long src = ei[e];
    const long long dst = ei[(long long)E + e];
    const float4 vv = *(const float4*)(v + dst * EMBED + r4);
    float* ap = accum + src * EMBED + r4;
    atomicAdd(ap + 0, vv.x * attn);
    atomicAdd(ap + 1, vv.y * attn);
    atomicAdd(ap + 2, vv.z * attn);
    atomicAdd(ap + 3, vv.w * attn);
}

// ---------------------------------------------------------------------------
extern "C" void kernel_launch(void* const* d_in, const int* in_sizes, int n_in,
                              void* d_out, int out_size, void* d_ws, size_t ws_size,
                              hipStream_t stream) {
    const float*     X  = (const float*)d_in[0];
    const long long* ei = (const long long*)d_in[1];         // jnp.int64
    const float*     Wq = (const float*)d_in[2];
    const float*     Wk = (const float*)d_in[3];
    const float*     Wv = (const float*)d_in[4];
    const float*     Wo = (const float*)d_in[5];
    const float*     bo = (const float*)d_in[6];
    float*           out = (float*)d_out;

    const int N = in_sizes[0] / EMBED;
    const int E = in_sizes[1] / 2;
    const size_t nd = (size_t)N * EMBED;

    char* ws = (char*)d_ws;
    float*    q      = (float*)ws;    ws += nd * sizeof(float);
    float*    kbuf   = (float*)ws;    ws += nd * sizeof(float);
    float*    v      = (float*)ws;    ws += nd * sizeof(float);
    float*    accum  = (float*)ws;    ws += nd * sizeof(float);
    float*    scores = (float*)ws;    ws += (size_t)E * NHEADS * sizeof(float);
    unsigned* hmaxo  = (unsigned*)ws; ws += NHEADS * sizeof(unsigned);
    float*    hsum   = (float*)ws;

    const int rowBlocks = (N + 15) / 16;
    const size_t shbytes = (size_t)(EMBED * LSTRIDE + 16 * LSTRIDE) * sizeof(float); // ~76KB

    { // init accumulator + head stats
        long long n = (long long)nd;
        int blocks = (int)((n + 255) / 256);
        init_kernel<<<blocks, 256, 0, stream>>>(accum, hmaxo, hsum, n);
    }
    { // QKV projections (WMMA f32, LDS-staged operands)
        qkv_kernel<<<3 * rowBlocks, 256, shbytes, stream>>>(X, Wq, Wk, Wv, q, kbuf, v, N, rowBlocks);
    }
    { // per-edge scores + softmax stats
        long long tcnt = (long long)E * NHEADS;
        int blocks = (int)((tcnt + 255) / 256);
        score_kernel<<<blocks, 256, 0, stream>>>(q, kbuf, ei, scores, hmaxo, E);
        expsum_kernel<<<blocks, 256, 0, stream>>>(scores, hmaxo, hsum, E);
    }
    { // weighted scatter-add
        long long tcnt = (long long)E * (EMBED / 4);
        int blocks = (int)((tcnt + 255) / 256);
        scatter_kernel<<<blocks, 256, 0, stream>>>(scores, hsum, v, ei, accum, E);
    }
    { // output projection (WMMA f32) + bias
        outproj_kernel<<<rowBlocks, 256, shbytes, stream>>>(accum, Wo, bo, out, N);
    }
}